// Dispersion_29678224015826
// MI455X (gfx1250) — compile-verified
//
#include <hip/hip_runtime.h>
#include <hip/hip_bf16.h>
#include <stdint.h>

// ---------------------------------------------------------------------------
// Dispersion (Tang-Toennies damped C6) pair energies.
// HBM-streaming bound: NT b128 loads of pairs, NT b128 stores of energies,
// TDM (tensor_load_to_lds) DMA of the 8x8 c6/b tables into LDS.
// ---------------------------------------------------------------------------

typedef __attribute__((ext_vector_type(4))) unsigned int u32x4;
typedef __attribute__((ext_vector_type(8))) int          i32x8;
typedef __attribute__((ext_vector_type(4))) int          i32x4;
// Native clang vectors for nontemporal builtins (HIP_vector_type is rejected).
typedef __attribute__((ext_vector_type(4))) int          vi4;
typedef __attribute__((ext_vector_type(4))) float        vf4;

#if __has_include(<hip/amd_detail/amd_gfx1250_TDM.h>)
// amdgpu-toolchain (clang-23 / therock-10.0): 6-arg builtin
#define TDM_LOAD_TO_LDS(g0, g1, g2, g3) \
    __builtin_amdgcn_tensor_load_to_lds((g0), (g1), (g2), (g3), (i32x8){0,0,0,0,0,0,0,0}, 0)
#else
// ROCm 7.2 (clang-22): 5-arg builtin
#define TDM_LOAD_TO_LDS(g0, g1, g2, g3) \
    __builtin_amdgcn_tensor_load_to_lds((g0), (g1), (g2), (g3), 0)
#endif

// Low 32 bits of a generic pointer to __shared__ == byte offset inside LDS.
__device__ __forceinline__ unsigned lds_byte_offset(const void* p) {
    return (unsigned)(unsigned long long)(uintptr_t)p;
}

// Build a D# for a 2-D fp32 tensor tile load (tile == whole tensor here) and
// issue TENSOR_LOAD_TO_LDS. Descriptor layout per CDNA5 ISA ch.8.
__device__ __forceinline__ void tdm_load_2d_f32(const float* __restrict__ gsrc,
                                                unsigned lds_off,
                                                unsigned dim0, unsigned dim1,
                                                unsigned stride0) {
    unsigned long long ga = (unsigned long long)(uintptr_t)gsrc;
    u32x4 g0;
    g0.x = 1u;                                   // count=1 (valid), user mode
    g0.y = lds_off;                              // lds_addr (bytes)
    g0.z = (unsigned)(ga & 0xFFFFFFFFull);       // global_addr[31:0]
    g0.w = (unsigned)((ga >> 32) & 0x01FFFFFFull) | (2u << 30); // addr[56:32] | type=2

    i32x8 g1;
    g1[0] = (int)(2u << 16);                     // wg_mask=0, data_size=2 (4B)
    g1[1] = (int)((dim0 & 0xFFFFu) << 16);       // tensor_dim0[15:0]
    g1[2] = (int)(((dim0 >> 16) & 0xFFFFu) | ((dim1 & 0xFFFFu) << 16));
    g1[3] = (int)(((dim1 >> 16) & 0xFFFFu) | ((dim0 & 0xFFFFu) << 16)); // tile_dim0=dim0
    g1[4] = (int)(dim1 & 0xFFFFu);               // tile_dim1=dim1, tile_dim2=0
    g1[5] = (int)stride0;                        // tensor_dim0_stride[31:0]
    g1[6] = 0;                                   // stride0[47:32] | stride1[15:0]
    g1[7] = 0;                                   // stride1[47:16]

    i32x4 gz = {0, 0, 0, 0};                     // groups 2/3 unused (2-D tensor)
    TDM_LOAD_TO_LDS(g0, g1, gz, gz);
}

// ---------------------------------------------------------------------------
// Setup kernel: 3x3 box inverse into workspace (reference does inv(box)).
// ---------------------------------------------------------------------------
__global__ void Dispersion_invbox_kernel(const float* __restrict__ box,
                                         float* __restrict__ inv) {
    if (blockIdx.x == 0 && threadIdx.x == 0) {
        float a = box[0], b = box[1], c = box[2];
        float d = box[3], e = box[4], f = box[5];
        float g = box[6], h = box[7], i = box[8];
        float C00 =  (e * i - f * h), C01 = -(d * i - f * g), C02 =  (d * h - e * g);
        float C10 = -(b * i - c * h), C11 =  (a * i - c * g), C12 = -(a * h - b * g);
        float C20 =  (b * f - c * e), C21 = -(a * f - c * d), C22 =  (a * e - b * d);
        float rd = 1.0f / (a * C00 + b * C01 + c * C02);
        inv[0] = C00 * rd; inv[1] = C10 * rd; inv[2] = C20 * rd;
        inv[3] = C01 * rd; inv[4] = C11 * rd; inv[5] = C21 * rd;
        inv[6] = C02 * rd; inv[7] = C12 * rd; inv[8] = C22 * rd;
    }
}

// ---------------------------------------------------------------------------
// Main pair-energy kernel.
// ---------------------------------------------------------------------------
struct DispCtx {
    const float* coords;
    const int*   types;
    float ib[9];   // inv(box), row-major
    float bx[9];   // box, row-major
    float cut2;
};

__device__ __forceinline__ float pair_energy(const DispCtx& c, int ai, int aj,
                                             const float* __restrict__ s_c6,
                                             const float* __restrict__ s_b) {
    const float* pi = c.coords + 3 * (long)ai;
    const float* pj = c.coords + 3 * (long)aj;
    float dx = pj[0] - pi[0];
    float dy = pj[1] - pi[1];
    float dz = pj[2] - pi[2];
    // s = dr @ inv_box  (row vector * matrix)
    float sx = fmaf(dx, c.ib[0], fmaf(dy, c.ib[3], dz * c.ib[6]));
    float sy = fmaf(dx, c.ib[1], fmaf(dy, c.ib[4], dz * c.ib[7]));
    float sz = fmaf(dx, c.ib[2], fmaf(dy, c.ib[5], dz * c.ib[8]));
    sx = rintf(sx); sy = rintf(sy); sz = rintf(sz);   // round-half-even
    // dr -= round(s) @ box
    dx -= fmaf(sx, c.bx[0], fmaf(sy, c.bx[3], sz * c.bx[6]));
    dy -= fmaf(sx, c.bx[1], fmaf(sy, c.bx[4], sz * c.bx[7]));
    dz -= fmaf(sx, c.bx[2], fmaf(sy, c.bx[5], sz * c.bx[8]));
    float r2 = fmaf(dx, dx, fmaf(dy, dy, dz * dz));

    int ti = c.types[ai];
    int tj = c.types[aj];
    int idx = ti * 8 + tj;
    float c6ij = s_c6[idx];      // ds_load_b32
    float bij  = s_b[idx];       // ds_load_b32

    float r = sqrtf(r2);
    float u = bij * r;
    // Horner FMA chain for Tang-Toennies order-6 polynomial
    float t = fmaf(u, 0.16666667f, 1.0f);
    t = fmaf(u * 0.2f,        t, 1.0f);
    t = fmaf(u * 0.25f,       t, 1.0f);
    t = fmaf(u * 0.33333334f, t, 1.0f);
    t = fmaf(u * 0.5f,        t, 1.0f);
    t = fmaf(u,               t, 1.0f);
    float f6 = 1.0f - __expf(-u) * t;            // v_exp_f32
    float r6 = r2 * r2 * r2;
    float e = -(c6ij * f6) / r6;
    return (r2 <= c.cut2) ? e : 0.0f;
}

__global__ __launch_bounds__(256, 4)
void Dispersion_29678224015826_kernel(const float* __restrict__ coords,
                                      const int*   __restrict__ pairs,
                                      const float* __restrict__ box,
                                      const float* __restrict__ c6,
                                      const float* __restrict__ bmat,
                                      const float* __restrict__ cutoff_p,
                                      const int*   __restrict__ types,
                                      const float* __restrict__ invbox,
                                      float*       __restrict__ out,
                                      int n_pairs) {
    __shared__ float s_c6[64];
    __shared__ float s_b[64];

    // TDM DMA of both 8x8 fp32 tables into LDS (per-wave issue; EXEC ignored
    // by tensor ops, redundant idempotent writes are harmless).
    tdm_load_2d_f32(c6,   lds_byte_offset(s_c6), 8u, 8u, 8u);
    tdm_load_2d_f32(bmat, lds_byte_offset(s_b),  8u, 8u, 8u);
    __builtin_amdgcn_s_wait_tensorcnt(0);
    __syncthreads();

    DispCtx ctx;
    ctx.coords = coords;
    ctx.types  = types;
#pragma unroll
    for (int k = 0; k < 9; ++k) { ctx.ib[k] = invbox[k]; ctx.bx[k] = box[k]; }
    float cut = cutoff_p[0];
    ctx.cut2 = cut * cut;

    long tid  = (long)blockIdx.x * blockDim.x + threadIdx.x;
    long base = tid * 4;

    if (base + 3 < (long)n_pairs) {
        // 4 pairs per lane: two NT b128 loads of indices, one NT b128 store.
        const vi4* p4 = (const vi4*)pairs;
        vi4 pa = __builtin_nontemporal_load(&p4[2 * tid]);
        vi4 pb = __builtin_nontemporal_load(&p4[2 * tid + 1]);
        vf4 res;
        res.x = pair_energy(ctx, pa.x, pa.y, s_c6, s_b);
        res.y = pair_energy(ctx, pa.z, pa.w, s_c6, s_b);
        res.z = pair_energy(ctx, pb.x, pb.y, s_c6, s_b);
        res.w = pair_energy(ctx, pb.z, pb.w, s_c6, s_b);
        __builtin_nontemporal_store(res, (vf4*)out + tid);
    } else {
        for (long p = base; p < (long)n_pairs; ++p) {
            int ai = pairs[2 * p];
            int aj = pairs[2 * p + 1];
            out[p] = pair_energy(ctx, ai, aj, s_c6, s_b);
        }
    }
}

// ---------------------------------------------------------------------------
// Launcher. Inputs: 0=coords 1=pairs 2=box 3=c6 4=b 5=cutoff 6=atom_types.
// ---------------------------------------------------------------------------
extern "C" void kernel_launch(void* const* d_in, const int* in_sizes, int n_in,
                              void* d_out, int out_size, void* d_ws, size_t ws_size,
                              hipStream_t stream) {
    const float* coords   = (const float*)d_in[0];
    const int*   pairs    = (const int*)d_in[1];
    const float* box      = (const float*)d_in[2];
    const float* c6       = (const float*)d_in[3];
    const float* bmat     = (const float*)d_in[4];
    const float* cutoff_p = (const float*)d_in[5];
    const int*   types    = (const int*)d_in[6];
    float*       out      = (float*)d_out;
    float*       invbox   = (float*)d_ws;   // 9 floats of scratch

    int n_pairs = out_size;

    Dispersion_invbox_kernel<<<1, 32, 0, stream>>>(box, invbox);

    long n_threads = ((long)n_pairs + 3) / 4;
    int  n_blocks  = (int)((n_threads + 255) / 256);
    if (n_blocks < 1) n_blocks = 1;
    Dispersion_29678224015826_kernel<<<n_blocks, 256, 0, stream>>>(
        coords, pairs, box, c6, bmat, cutoff_p, types, invbox, out, n_pairs);
}